// DistributionLossWithLabel_v2_17008070493058
// MI455X (gfx1250) — compile-verified
//
#include <hip/hip_runtime.h>
#include <hip/hip_bf16.h>

#define B_DIM 4096
#define D_DIM 1024
#define TILE  128
#define KB    32
#define LDSS  36   // padded LDS row stride (floats): multiple of 4 for float4 stores, odd*4 for bank spread

typedef __attribute__((ext_vector_type(2))) float v2f;
typedef __attribute__((ext_vector_type(8))) float v8f;

// ---------------------------------------------------------------------------
// Kernel 1: per-row stats.  One block per row.
//   logq[i,d] = log(q[i,d])          (stored: reused B/TILE times by the GEMM)
//   p_ent[i]  = sum_d p*log p
//   kl_div[i] = mean_d p*(log p - log q)
//   pos_acc/sum_acc zero-init (re-zeroed every launch -> graph-replay safe)
// ---------------------------------------------------------------------------
__global__ __launch_bounds__(256) void rowstats_kernel(
    const float* __restrict__ q, const float* __restrict__ p,
    float* __restrict__ logq, float* __restrict__ p_ent,
    float* __restrict__ kl_div, float* __restrict__ pos_acc,
    float* __restrict__ sum_acc)
{
    const int row = blockIdx.x;
    const int t   = threadIdx.x;
    const float* qr = q + (size_t)row * D_DIM;
    const float* pr = p + (size_t)row * D_DIM;
    float*       lr = logq + (size_t)row * D_DIM;

    float s_ent = 0.f, s_kl = 0.f;
    #pragma unroll
    for (int d = t; d < D_DIM; d += 256) {
        float qv = qr[d], pv = pr[d];
        float lq = __builtin_logf(qv);
        float lp = __builtin_logf(pv);
        lr[d] = lq;
        s_ent += pv * lp;
        s_kl  += pv * (lp - lq);
    }
    __shared__ float red0[256], red1[256];
    red0[t] = s_ent; red1[t] = s_kl;
    __syncthreads();
    for (int off = 128; off > 0; off >>= 1) {
        if (t < off) { red0[t] += red0[t + off]; red1[t] += red1[t + off]; }
        __syncthreads();
    }
    if (t == 0) {
        p_ent[row]   = red0[0];
        kl_div[row]  = red1[0] * (1.0f / D_DIM);
        pos_acc[row] = 0.f;
        sum_acc[row] = 0.f;
    }
}

// ---------------------------------------------------------------------------
// Kernel 2: fused GEMM + KL epilogue.
//   cross[i,j] = sum_d logq[i,d]*p[j,d]   via V_WMMA_F32_16X16X4_F32
//   kl = (p_ent[j]-cross)/D ; pos_acc[i] += sum_j kl*L[i,j] ; sum_acc[i] += sum_j kl
// Block = 256 thr (8 waves), C-tile 128x128; wave tile 64x32 = 4x2 subtiles.
// ---------------------------------------------------------------------------
__global__ __launch_bounds__(256) void klgemm_kernel(
    const float* __restrict__ logq, const float* __restrict__ p,
    const float* __restrict__ labels, const float* __restrict__ p_ent,
    float* __restrict__ pos_acc, float* __restrict__ sum_acc)
{
    __shared__ float As[TILE][LDSS];   // [m][k]
    __shared__ float Bs[TILE][LDSS];   // [n][k]

    const int t    = threadIdx.x;
    const int lane = t & 31;
    const int w    = t >> 5;
    const int wm   = (w & 1) * 64;          // wave M offset in tile
    const int wn   = (w >> 1) * 32;         // wave N offset in tile
    const int i0   = blockIdx.y * TILE;
    const int j0   = blockIdx.x * TILE;

    // ISA 16x4 f32 A/B fragment addressing: lane L -> row (L&15), k-pair (L>>4)*2
    const int lm = lane & 15;
    const int lk = (lane >> 4) * 2;

    v8f acc[4][2];
    const v8f vzero = {0.f, 0.f, 0.f, 0.f, 0.f, 0.f, 0.f, 0.f};
    #pragma unroll
    for (int sm = 0; sm < 4; ++sm)
        #pragma unroll
        for (int sn = 0; sn < 2; ++sn)
            acc[sm][sn] = vzero;

    for (int kb = 0; kb < D_DIM; kb += KB) {
        __syncthreads();   // previous iteration's LDS reads done
        // stage A (logq) and B (p) 128x32 fp32 tiles; 4 float4 each per thread
        #pragma unroll
        for (int it = 0; it < 4; ++it) {
            int idx = t + it * 256;
            int r   = idx >> 3;          // 0..127
            int c   = (idx & 7) * 4;     // 0..28
            float4 av = *(const float4*)(logq + (size_t)(i0 + r) * D_DIM + kb + c);
            float4 bv = *(const float4*)(p    + (size_t)(j0 + r) * D_DIM + kb + c);
            *(float4*)&As[r][c] = av;    // LDSS%4==0 -> 16B aligned
            *(float4*)&Bs[r][c] = bv;
        }
        __syncthreads();

        #pragma unroll
        for (int kk = 0; kk < KB; kk += 4) {
            v2f afrag[4], bfrag[2];
            #pragma unroll
            for (int sm = 0; sm < 4; ++sm) {
                const float* pa = &As[wm + sm * 16 + lm][kk + lk];
                afrag[sm] = (v2f){pa[0], pa[1]};
            }
            #pragma unroll
            for (int sn = 0; sn < 2; ++sn) {
                const float* pb = &Bs[wn + sn * 16 + lm][kk + lk];
                bfrag[sn] = (v2f){pb[0], pb[1]};
            }
            #pragma unroll
            for (int sm = 0; sm < 4; ++sm)
                #pragma unroll
                for (int sn = 0; sn < 2; ++sn)
                    acc[sm][sn] = __builtin_amdgcn_wmma_f32_16x16x4_f32(
                        false, afrag[sm], false, bfrag[sn],
                        (short)0, acc[sm][sn], false, false);
        }
    }

    // ---- epilogue: kl_dis, labels weighting, per-row reduction, atomics ----
    const float invD = 1.0f / D_DIM;
    const float pe0 = p_ent[j0 + wn + lm];        // sn = 0 column entropy
    const float pe1 = p_ent[j0 + wn + 16 + lm];   // sn = 1
    const int   hi8 = (lane >> 4) * 8;            // C/D layout: lanes 16-31 hold M+8

    #pragma unroll
    for (int sm = 0; sm < 4; ++sm) {
        const int i_base = i0 + wm + sm * 16 + hi8;
        #pragma unroll
        for (int r = 0; r < 8; ++r) {
            const int i = i_base + r;
            const size_t lrow = (size_t)i * B_DIM + j0 + wn + lm;
            float kl0 = (pe0 - acc[sm][0][r]) * invD;
            float kl1 = (pe1 - acc[sm][1][r]) * invD;
            float lab0 = labels[lrow];
            float lab1 = labels[lrow + 16];
            float pos = kl0 * lab0 + kl1 * lab1;
            float tot = kl0 + kl1;
            // reduce across the 16 lanes of each half-wave (N direction)
            #pragma unroll
            for (int off = 8; off >= 1; off >>= 1) {
                pos += __shfl_xor(pos, off, 32);
                tot += __shfl_xor(tot, off, 32);
            }
            if (lm == 0) {
                atomicAdd(&pos_acc[i], pos);
                atomicAdd(&sum_acc[i], tot);
            }
        }
    }
}

// ---------------------------------------------------------------------------
// Kernel 3: final scalar = sum_i (kl_div[i] + pos)/(tot - pos)
// ---------------------------------------------------------------------------
__global__ __launch_bounds__(256) void finalize_kernel(
    const float* __restrict__ kl_div, const float* __restrict__ pos_acc,
    const float* __restrict__ sum_acc, float* __restrict__ out)
{
    const int t = threadIdx.x;
    float s = 0.f;
    for (int i = t; i < B_DIM; i += 256) {
        float pos = pos_acc[i];
        float tot = sum_acc[i];
        s += (kl_div[i] + pos) / (tot - pos);
    }
    __shared__ float red[256];
    red[t] = s;
    __syncthreads();
    for (int off = 128; off > 0; off >>= 1) {
        if (t < off) red[t] += red[t + off];
        __syncthreads();
    }
    if (t == 0) out[0] = red[0];
}

extern "C" void kernel_launch(void* const* d_in, const int* in_sizes, int n_in,
                              void* d_out, int out_size, void* d_ws, size_t ws_size,
                              hipStream_t stream) {
    const float* q      = (const float*)d_in[0];
    const float* p      = (const float*)d_in[1];
    const float* labels = (const float*)d_in[2];

    float* logq    = (float*)d_ws;                       // B*D fp32 = 16 MB
    float* p_ent   = logq + (size_t)B_DIM * D_DIM;       // B
    float* kl_div  = p_ent + B_DIM;                      // B
    float* pos_acc = kl_div + B_DIM;                     // B
    float* sum_acc = pos_acc + B_DIM;                    // B

    rowstats_kernel<<<B_DIM, 256, 0, stream>>>(q, p, logq, p_ent, kl_div,
                                               pos_acc, sum_acc);
    dim3 grid(B_DIM / TILE, B_DIM / TILE);
    klgemm_kernel<<<grid, 256, 0, stream>>>(logq, p, labels, p_ent,
                                            pos_acc, sum_acc);
    finalize_kernel<<<1, 256, 0, stream>>>(kl_div, pos_acc, sum_acc,
                                           (float*)d_out);
}